// DoRALinear_14869176779242
// MI455X (gfx1250) — compile-verified
//
#include <hip/hip_runtime.h>
#include <math.h>

// ---- CDNA5 WMMA types -------------------------------------------------------
typedef __attribute__((ext_vector_type(2))) float v2f;  // A/B frag: 16x4 / 4x16 f32
typedef __attribute__((ext_vector_type(8))) float v8f;  // C/D frag: 16x16 f32

// D = A(16x4) * B(4x16) + C, full f32.
// (neg_a, A, neg_b, B, c_mod, C, reuse_a, reuse_b)
static __device__ __forceinline__ v8f wmma_f32(v2f a, v2f b, v8f c) {
  return __builtin_amdgcn_wmma_f32_16x16x4_f32(false, a, false, b, (short)0, c,
                                               false, false);
}

// Problem constants (from reference setup_inputs)
#define MROWS 8192   // B*S
#define DIN   4096
#define DOUT  4096
#define RANK  64
#define SCALING 2.0f
#define EPS 1e-8f
#define KCHUNKS 8            // split-K factor for kernel 2
#define KC (DIN / KCHUNKS)   // 512

// ---------------------------------------------------------------------------
// Kernel 1: mag_scale[o] = magnitude[o] / (||W[o,:] + 2 * (B@A)[o,:]||_2 + eps)
// One 256-thread block per output row. W streamed once; lora_A (1MB) from L2.
// ---------------------------------------------------------------------------
__global__ void __launch_bounds__(256)
dora_magscale(const float* __restrict__ W, const float* __restrict__ lA,
              const float* __restrict__ lB, const float* __restrict__ mag,
              float* __restrict__ mag_scale) {
  __shared__ float bs[RANK];
  __shared__ float red[256];
  const int o = blockIdx.x;
  const int t = threadIdx.x;
  if (t < RANK) bs[t] = lB[(size_t)o * RANK + t];
  __syncthreads();

  float sum = 0.0f;
  for (int d = t; d < DIN; d += 256) {
    float acc = 0.0f;
#pragma unroll 8
    for (int r = 0; r < RANK; ++r) acc = fmaf(bs[r], lA[(size_t)r * DIN + d], acc);
    float u = W[(size_t)o * DIN + d] + SCALING * acc;
    sum = fmaf(u, u, sum);
  }
  red[t] = sum;
  __syncthreads();
  for (int s = 128; s > 0; s >>= 1) {
    if (t < s) red[t] += red[t + s];
    __syncthreads();
  }
  if (t == 0) mag_scale[o] = mag[o] / (sqrtf(red[0]) + EPS);
}

// ---------------------------------------------------------------------------
// Kernel 2: split-K partials of x @ A^T.
// GEMM M=8192, N=64, K=4096 with v_wmma_f32_16x16x4_f32.
// Wave w: mtile = w>>3 (16 rows), chunk = w&7 (512 K-values).
// Frag addressing (wave32):
//   A 16x4: lanes L / L+16 -> row m0+L, float2 at K kk + 2*(lane>>4)
//   B 4x16: lanes L / L+16 -> col n0+L, float2 at K kk + 2*(lane>>4)
// part[chunk][m][n] = sum over this chunk's K range (unscaled).
// ---------------------------------------------------------------------------
__global__ void __launch_bounds__(256)
lora_xa_wmma(const float* __restrict__ X, const float* __restrict__ lA,
             float* __restrict__ Part) {
  const int wave = (blockIdx.x * blockDim.x + threadIdx.x) >> 5;  // 0..4095
  const int lane = threadIdx.x & 31;
  const int chunk = wave & (KCHUNKS - 1);
  const int m0 = (wave >> 3) * 16;
  const int l15 = lane & 15;
  const int half = lane >> 4;
  const int k0 = chunk * KC;

  const float* xrow = X + (size_t)(m0 + l15) * DIN + k0 + 2 * half;
  const float* arow0 = lA + (size_t)l15 * DIN + k0 + 2 * half;  // n-tile 0 base

  v8f acc[4] = {v8f{}, v8f{}, v8f{}, v8f{}};

  for (int kk = 0; kk < KC; kk += 16) {
    __builtin_prefetch(xrow + kk + 256, 0, 3);  // global_prefetch ~1KB ahead
#pragma unroll
    for (int k4 = 0; k4 < 16; k4 += 4) {
      const int k = kk + k4;
      v2f af = *(const v2f*)(xrow + k);
#pragma unroll
      for (int nt = 0; nt < 4; ++nt) {
        v2f bf = *(const v2f*)(arow0 + (size_t)(nt * 16) * DIN + k);
        acc[nt] = wmma_f32(af, bf, acc[nt]);
      }
    }
  }

  // C layout: VGPR i, lanes 0-15 -> (M=i, N=lane); lanes 16-31 -> (M=i+8, N=lane-16)
  float* pdst = Part + (size_t)chunk * MROWS * RANK;
#pragma unroll
  for (int nt = 0; nt < 4; ++nt) {
#pragma unroll
    for (int i = 0; i < 8; ++i) {
      size_t row = (size_t)(m0 + i + 8 * half);
      pdst[row * RANK + nt * 16 + l15] = acc[nt][i];
    }
  }
}

// ---------------------------------------------------------------------------
// Kernel 2b: xa_s[i] = SCALING * sum_c part[c][i]   (i over MROWS*RANK)
// ---------------------------------------------------------------------------
__global__ void __launch_bounds__(256)
lora_xa_reduce(const float* __restrict__ Part, float* __restrict__ XAs) {
  const size_t i = (size_t)blockIdx.x * blockDim.x + threadIdx.x;
  float s = 0.0f;
#pragma unroll
  for (int c = 0; c < KCHUNKS; ++c) s += Part[(size_t)c * MROWS * RANK + i];
  XAs[i] = SCALING * s;
}

// ---------------------------------------------------------------------------
// Kernel 3: out[m, n] = (base_output[m,n] + sum_r xa_s[m,r]*lora_B[n,r]) * ms[n]
// GEMM M=8192, N=4096, K=64 + fused DoRA epilogue.
// One wave per 16(M) x 64(N) tile strip (32768 waves).
// ---------------------------------------------------------------------------
__global__ void __launch_bounds__(256)
lora_out_wmma(const float* __restrict__ XAs, const float* __restrict__ lB,
              const float* __restrict__ baseO, const float* __restrict__ ms,
              float* __restrict__ Out) {
  const int wave = (blockIdx.x * blockDim.x + threadIdx.x) >> 5;  // 0..32767
  const int lane = threadIdx.x & 31;
  const int mtile = wave >> 6;    // 0..511
  const int nstrip = wave & 63;   // 0..63 (each covers 64 cols)
  const int m0 = mtile * 16;
  const int l15 = lane & 15;
  const int half = lane >> 4;

  // Warm the epilogue tile: rows m0..m0+15, 256B each (two 128B lines / row).
  __builtin_prefetch(baseO + (size_t)(m0 + l15) * DOUT + nstrip * 64 + half * 32,
                     0, 3);

  const float* arow = XAs + (size_t)(m0 + l15) * RANK + 2 * half;
  const float* brow0 = lB + (size_t)(nstrip * 64 + l15) * RANK + 2 * half;

  v8f acc[4] = {v8f{}, v8f{}, v8f{}, v8f{}};

#pragma unroll
  for (int k = 0; k < RANK; k += 4) {
    v2f af = *(const v2f*)(arow + k);
#pragma unroll
    for (int nt = 0; nt < 4; ++nt) {
      v2f bf = *(const v2f*)(brow0 + (size_t)(nt * 16) * RANK + k);
      acc[nt] = wmma_f32(af, bf, acc[nt]);
    }
  }

#pragma unroll
  for (int nt = 0; nt < 4; ++nt) {
    const int col = nstrip * 64 + nt * 16 + l15;
    const float m = ms[col];
#pragma unroll
    for (int i = 0; i < 8; ++i) {
      size_t idx = (size_t)(m0 + i + 8 * half) * DOUT + col;
      Out[idx] = (baseO[idx] + acc[nt][i]) * m;
    }
  }
}

// ---------------------------------------------------------------------------
extern "C" void kernel_launch(void* const* d_in, const int* in_sizes, int n_in,
                              void* d_out, int out_size, void* d_ws, size_t ws_size,
                              hipStream_t stream) {
  const float* x      = (const float*)d_in[0];  // [2,4096,4096]
  const float* baseO  = (const float*)d_in[1];  // [2,4096,4096]
  const float* baseW  = (const float*)d_in[2];  // [4096,4096]
  const float* lA     = (const float*)d_in[3];  // [64,4096]
  const float* lB     = (const float*)d_in[4];  // [4096,64]
  const float* mag    = (const float*)d_in[5];  // [4096]
  float* out = (float*)d_out;

  // workspace layout (floats):
  //   [0, DOUT)                                   mag_scale
  //   [DOUT, DOUT + KCHUNKS*MROWS*RANK)           split-K partials (16MB)
  //   [.., .. + MROWS*RANK)                       reduced, scaled xa (2MB)
  float* ws_magscale = (float*)d_ws;
  float* ws_part = ws_magscale + DOUT;
  float* ws_xas = ws_part + (size_t)KCHUNKS * MROWS * RANK;

  // K1: per-row updated-weight norm -> mag_scale  (4096 blocks x 256)
  dora_magscale<<<DOUT, 256, 0, stream>>>(baseW, lA, lB, mag, ws_magscale);

  // K2: split-K partials of x @ A^T  (4096 waves -> 512 blocks x 256)
  lora_xa_wmma<<<(MROWS / 16) * KCHUNKS * 32 / 256, 256, 0, stream>>>(x, lA,
                                                                      ws_part);

  // K2b: reduce partials, fold SCALING  (2048 blocks x 256)
  lora_xa_reduce<<<(MROWS * RANK) / 256, 256, 0, stream>>>(ws_part, ws_xas);

  // K3: fused GEMM + DoRA epilogue (32768 waves -> 4096 blocks x 256)
  lora_out_wmma<<<(MROWS / 16) * (DOUT / 64) * 32 / 256, 256, 0, stream>>>(
      ws_xas, lB, baseO, ws_magscale, out);
}